// DropoutHyperbolic_7370163880195
// MI455X (gfx1250) — compile-verified
//
#include <hip/hip_runtime.h>
#include <math.h>

// ---------------------------------------------------------------------------
// Hyperbolic dropout (logmap @ origin -> dropout -> expmap @ origin), fp32.
// N=16384 rows, D=512. Memory-bound: ~100 MB traffic -> ~4.3 us @ 23.3 TB/s.
// One wave32 per row; row data staged via CDNA5 async global->LDS (ASYNCcnt).
// ---------------------------------------------------------------------------

#define RATE       0.1f
#define CLIP_MAG   8.0f
#define EPS_C      1e-4f
#define ACOSH_MIN  1.0001f

constexpr int D               = 512;
constexpr int WAVES_PER_BLOCK = 8;
constexpr int THREADS         = WAVES_PER_BLOCK * 32;   // 256
constexpr int F4_PER_ROW      = D / 4;                  // 128 float4 per row
constexpr int F4_PER_LANE     = F4_PER_ROW / 32;        // 4  float4 per lane

__global__ __launch_bounds__(THREADS)
void hyper_dropout_kernel(const float* __restrict__ vecs,
                          const float* __restrict__ curvature,
                          const float* __restrict__ mask_u,
                          float* __restrict__ out,
                          int nrows)
{
    // Per block: 8 waves * (128 vec + 128 mask) float4 = 2048 * 16B = 32 KB LDS.
    __shared__ float4 stage[WAVES_PER_BLOCK * 2 * F4_PER_ROW];

    const int lane = threadIdx.x & 31;
    const int wave = threadIdx.x >> 5;
    const int row  = blockIdx.x * WAVES_PER_BLOCK + wave;
    if (row >= nrows) return;   // whole-wave uniform exit

    const float c      = curvature[0];
    const float sqrt_c = sqrtf(c);

    const float4* __restrict__ vrow = reinterpret_cast<const float4*>(vecs   + (size_t)row * D);
    const float4* __restrict__ mrow = reinterpret_cast<const float4*>(mask_u + (size_t)row * D);
    float4*       __restrict__ orow = reinterpret_cast<float4*>      (out    + (size_t)row * D);

    float4* shv = &stage[wave * 2 * F4_PER_ROW];
    float4* shm = shv + F4_PER_ROW;

    // ---- CDNA5 async global->LDS staging of this wave's row (vecs + mask) ----
    // Per-lane gather: each lane copies 16B; 4 b128 ops cover one 2KB row.
    // LDS address operand = low 32 bits of the generic shared pointer
    // (flat->LDS truncation, ISA sec 10.2).
#pragma unroll
    for (int q = 0; q < F4_PER_LANE; ++q) {
        const int idx = lane + 32 * q;
        uint64_t gv = (uint64_t)(uintptr_t)(vrow + idx);
        uint64_t gm = (uint64_t)(uintptr_t)(mrow + idx);
        uint32_t lv = (uint32_t)(uintptr_t)(shv + idx);
        uint32_t lm = (uint32_t)(uintptr_t)(shm + idx);
        asm volatile("global_load_async_to_lds_b128 %0, %1, off"
                     :: "v"(lv), "v"(gv) : "memory");
        asm volatile("global_load_async_to_lds_b128 %0, %1, off"
                     :: "v"(lm), "v"(gm) : "memory");
    }
    asm volatile("s_wait_asynccnt 0" ::: "memory");

    // ---- pull row into registers (16 + 16 floats per lane) ----
    float4 v[F4_PER_LANE], m[F4_PER_LANE];
#pragma unroll
    for (int q = 0; q < F4_PER_LANE; ++q) {
        v[q] = shv[lane + 32 * q];
        m[q] = shm[lane + 32 * q];
    }

    // v[0] of the row is discarded by the projection; zeroing it makes every
    // downstream formula (s2, kq, output) correct, and lane 0 patches out[0].
    if (lane == 0) v[0].x = 0.0f;

    // ---- fused reductions: s2 = sum v[1:]^2 ; kq = sum keep * v[1:]^2 ----
    float s2 = 0.0f, kq = 0.0f;
#pragma unroll
    for (int q = 0; q < F4_PER_LANE; ++q) {
        float x;
        x = v[q].x; s2 += x * x; if (m[q].x >= RATE) kq += x * x;
        x = v[q].y; s2 += x * x; if (m[q].y >= RATE) kq += x * x;
        x = v[q].z; s2 += x * x; if (m[q].z >= RATE) kq += x * x;
        x = v[q].w; s2 += x * x; if (m[q].w >= RATE) kq += x * x;
    }
#pragma unroll
    for (int off = 16; off > 0; off >>= 1) {
        s2 += __shfl_xor(s2, off, 32);
        kq += __shfl_xor(kq, off, 32);
    }

    // ---- per-row scalars (uniform across the wave) ----
    const float v0p    = sqrtf(c + s2);                         // projected first coord
    const float targ   = fmaxf(v0p / sqrt_c - EPS_C, ACOSH_MIN);
    const float dist   = sqrt_c * acoshf(targ);
    const float norm_u = sqrtf(s2);                             // hyperbolic norm of unnorm
    const float scale1 = dist / (norm_u + EPS_C);               // logmap scale
    const float dscale = scale1 * (1.0f / (1.0f - RATE));       // + dropout 1/(1-rate)
    const float sd2    = dscale * dscale * kq;                  // ||dropped||^2
    const float snv    = sqrtf(sd2) / sqrt_c + EPS_C;
    const float snv_c  = fminf(fmaxf(snv, -CLIP_MAG), CLIP_MAG);
    const float sh     = sinhf(snv_c);
    const float ch     = coshf(snv_c);
    const float oscale = (sh / snv) * dscale;                   // applied to keep*v[j]

    // ---- elementwise output ----
#pragma unroll
    for (int q = 0; q < F4_PER_LANE; ++q) {
        float4 y;
        y.x = (m[q].x >= RATE ? oscale : 0.0f) * v[q].x;
        y.y = (m[q].y >= RATE ? oscale : 0.0f) * v[q].y;
        y.z = (m[q].z >= RATE ? oscale : 0.0f) * v[q].z;
        y.w = (m[q].w >= RATE ? oscale : 0.0f) * v[q].w;
        if (q == 0 && lane == 0) y.x = ch * sqrt_c;  // out[0] = cosh * sqrt(c)
        orow[lane + 32 * q] = y;
    }
}

extern "C" void kernel_launch(void* const* d_in, const int* in_sizes, int n_in,
                              void* d_out, int out_size, void* d_ws, size_t ws_size,
                              hipStream_t stream)
{
    (void)n_in; (void)out_size; (void)d_ws; (void)ws_size;
    const float* vecs = (const float*)d_in[0];   // [N, 512] fp32
    const float* curv = (const float*)d_in[1];   // [1]      fp32
    const float* mask = (const float*)d_in[2];   // [N, 512] fp32
    float*       out  = (float*)d_out;           // [N, 512] fp32

    const int nrows = in_sizes[0] / D;           // 16384
    const int grid  = (nrows + WAVES_PER_BLOCK - 1) / WAVES_PER_BLOCK;

    hipLaunchKernelGGL(hyper_dropout_kernel, dim3(grid), dim3(THREADS), 0, stream,
                       vecs, curv, mask, out, nrows);
}